// ChunkSum_87205015978274
// MI455X (gfx1250) — compile-verified
//
#include <hip/hip_runtime.h>
#include <hip/hip_bf16.h>

#define NBATCH   64
#define CH_PER_B 525      // 5*5*21 chunks per batch
#define M_CH     33600    // 64*525
#define NSORT    65536
#define PADKEY   0x7FFFFFFFu
#define VS0      40
#define VS1      40
#define VS2      200

typedef unsigned int u32x4 __attribute__((ext_vector_type(4)));
typedef int          i32x8 __attribute__((ext_vector_type(8)));
typedef int          i32x4 __attribute__((ext_vector_type(4)));

// ---- key for chunk instance i (matches _chunkify flatten order) ----------
__device__ __forceinline__ unsigned chunk_key(const int* __restrict__ loc, int i) {
    int b  = i / CH_PER_B, j = i % CH_PER_B;
    int i0 = j / 105, i1 = (j / 21) % 5, i2 = j % 21;
    int cg0 = loc[3 * b + 0] / 10 + i0;   // start/10 + idx  (loc >= 0)
    int cg1 = loc[3 * b + 1] / 10 + i1;
    int cg2 = loc[3 * b + 2] / 10 + i2;
    return ((unsigned)cg0 << 16) | ((unsigned)cg1 << 8) | (unsigned)cg2;
}

// ---- 1) keys + pad, zero contributor masks -------------------------------
__global__ void kInit(const int* __restrict__ loc, unsigned* __restrict__ keys,
                      unsigned* __restrict__ mask2) {
    int i = blockIdx.x * blockDim.x + threadIdx.x;
    if (i < NSORT) {
        keys[i] = (i < M_CH) ? chunk_key(loc, i) : PADKEY;
    } else {
        int j = i - NSORT;
        if (j < 2 * M_CH) mask2[j] = 0u;
    }
}

// ---- 2) bitonic sort of 65536 u32 keys -----------------------------------
__global__ void kSortLocal(unsigned* __restrict__ keys) {  // stages k=2..1024
    __shared__ unsigned s[1024];
    int t = threadIdx.x, base = blockIdx.x * 1024;
    s[t] = keys[base + t];
    s[t + 512] = keys[base + t + 512];
    __syncthreads();
    for (int k = 2; k <= 1024; k <<= 1) {
        for (int j = k >> 1; j > 0; j >>= 1) {
            int i = 2 * t - (t & (j - 1));
            unsigned a = s[i], b = s[i + j];
            bool dir = (((base + i) & k) == 0);
            if ((a > b) == dir) { s[i] = b; s[i + j] = a; }
            __syncthreads();
        }
    }
    keys[base + t] = s[t];
    keys[base + t + 512] = s[t + 512];
}

__global__ void kMergeGlobal(unsigned* __restrict__ keys, int k, int j) {
    int t = blockIdx.x * blockDim.x + threadIdx.x;  // 0..65535
    int ixj = t ^ j;
    if (ixj > t) {
        unsigned a = keys[t], b = keys[ixj];
        bool dir = ((t & k) == 0);
        if ((a > b) == dir) { keys[t] = b; keys[ixj] = a; }
    }
}

__global__ void kMergeShared(unsigned* __restrict__ keys, int k) {  // j=512..1
    __shared__ unsigned s[1024];
    int t = threadIdx.x, base = blockIdx.x * 1024;
    s[t] = keys[base + t];
    s[t + 512] = keys[base + t + 512];
    __syncthreads();
    for (int j = 512; j > 0; j >>= 1) {
        int i = 2 * t - (t & (j - 1));
        unsigned a = s[i], b = s[i + j];
        bool dir = (((base + i) & k) == 0);
        if ((a > b) == dir) { s[i] = b; s[i + j] = a; }
        __syncthreads();
    }
    keys[base + t] = s[t];
    keys[base + t + 512] = s[t + 512];
}

// ---- 3) dedup sorted keys -> compacted sorted-unique + count U -----------
__global__ void kDedup(const unsigned* __restrict__ skeys, unsigned* __restrict__ ukeys,
                       unsigned* __restrict__ Uptr) {
    __shared__ unsigned cnt[1024];
    __shared__ unsigned pre[1025];
    int t = threadIdx.x, base = t * 64;
    unsigned prev = (base == 0) ? PADKEY : skeys[base - 1];
    unsigned c = 0;
    for (int q = 0; q < 64; ++q) {
        unsigned key = skeys[base + q];
        if (key != PADKEY && key != prev) c++;
        prev = key;
    }
    cnt[t] = c;
    __syncthreads();
    if (t == 0) {
        unsigned run = 0;
        for (int q = 0; q < 1024; ++q) { pre[q] = run; run += cnt[q]; }
        pre[1024] = run;
    }
    __syncthreads();
    unsigned w = pre[t];
    prev = (base == 0) ? PADKEY : skeys[base - 1];
    for (int q = 0; q < 64; ++q) {
        unsigned key = skeys[base + q];
        if (key != PADKEY && key != prev) ukeys[w++] = key;
        prev = key;
    }
    if (t == 0) *Uptr = pre[1024];
}

// ---- 4) per-instance binary search -> set contributor-batch bit ----------
__global__ void kScatterMask(const int* __restrict__ loc, const unsigned* __restrict__ ukeys,
                             const unsigned* __restrict__ Uptr, unsigned* __restrict__ mask2) {
    int i = blockIdx.x * blockDim.x + threadIdx.x;
    if (i >= M_CH) return;
    unsigned key = chunk_key(loc, i);
    unsigned lo = 0, hi = *Uptr;
    while (lo < hi) {
        unsigned mid = (lo + hi) >> 1;
        if (ukeys[mid] < key) lo = mid + 1; else hi = mid;
    }
    int b = i / CH_PER_B;
    atomicOr(&mask2[2u * lo + (unsigned)(b >> 5)], 1u << (b & 31));
}

// ---- 5) main gather: one wave32 per output row, TDM tile loads -----------
__global__ void __launch_bounds__(32)
kGather(const float* __restrict__ data, const int* __restrict__ loc,
        const unsigned* __restrict__ ukeys, const unsigned* __restrict__ Uptr,
        const unsigned long long* __restrict__ mask, float* __restrict__ out,
        int* __restrict__ uloc) {
    __shared__ float tile[1000];
    int r = blockIdx.x;
    int lane = threadIdx.x;
    unsigned U = *Uptr;
    float acc[32];
#pragma unroll
    for (int k = 0; k < 32; ++k) acc[k] = 0.0f;

    int cg0 = 32767, cg1 = 255, cg2 = 255;  // decode of fill key 2^31-1
    if (r < (int)U) {
        unsigned key = ukeys[r];
        cg0 = (int)(key >> 16); cg1 = (int)((key >> 8) & 255); cg2 = (int)(key & 255);
        int gx = cg0 * 10, gy = cg1 * 10, gz = cg2 * 10;
        unsigned long long m = mask[r];
        while (m) {
            int b = __ffsll(m) - 1;
            m &= m - 1;
            int d0 = gx - loc[3 * b + 0];
            int d1 = gy - loc[3 * b + 1];
            int d2 = gz - loc[3 * b + 2];
            int s0 = d0 > 0 ? d0 : 0, s1 = d1 > 0 ? d1 : 0, s2 = d2 > 0 ? d2 : 0;
            int e0 = (d0 + 10 < VS0) ? d0 + 10 : VS0;
            int e1 = (d1 + 10 < VS1) ? d1 + 10 : VS1;
            int e2 = (d2 + 10 < VS2) ? d2 + 10 : VS2;
            int a0 = e0 - s0, a1 = e1 - s1, a2 = e2 - s2;
            if (a0 <= 0 || a1 <= 0 || a2 <= 0) continue;  // all-padding chunk
            int off0 = s0 - d0, off1 = s1 - d1, off2 = s2 - d2;

            // ---- Tensor DMA descriptor (CDNA5 ISA) ----
            unsigned long long gaddr = (unsigned long long)(uintptr_t)data +
                4ull * ((unsigned long long)b * (VS0 * VS1 * VS2) +
                        (unsigned long long)(s0 * (VS1 * VS2) + s1 * VS2 + s2));
            unsigned ldsa = (unsigned)(size_t)(&tile[0]);  // low 32b of flat LDS addr = byte offset
            u32x4 G0 = { 1u,                                   // count=1, user descriptor
                         ldsa,                                 // lds_addr
                         (unsigned)gaddr,                      // global_addr[31:0]
                         (unsigned)((gaddr >> 32) & 0x1FFFFFFull) | (2u << 30) }; // addr[56:32] | type=2
            i32x8 G1 = { (int)(2u << 16),                      // data_size = 4B, no multicast
                         (int)((unsigned)VS2 << 16),           // tensor_dim0[15:0] = 200
                         (int)((unsigned)VS1 << 16),           // tensor_dim1[15:0] = 40
                         (int)((unsigned)a2 << 16),            // tile_dim0 = z extent
                         (int)((unsigned)a1 | ((unsigned)a0 << 16)), // tile_dim1=y, tile_dim2=x
                         (int)(VS2),                           // tensor_dim0_stride = 200
                         (int)(8000u << 16),                   // tensor_dim1_stride[15:0] = 8000
                         0 };                                  // tensor_dim1_stride[47:16] = 0
            i32x4 G2 = { VS0, 0, 0, 0 };                       // tensor_dim2 = 40; 3-D tile only
            i32x4 G3 = { 0, 0, 0, 0 };
            i32x8 G4 = { 0, 0, 0, 0, 0, 0, 0, 0 };             // unused extension operand
            __builtin_amdgcn_tensor_load_to_lds(G0, G1, G2, G3, G4, 0);
            __builtin_amdgcn_s_wait_tensorcnt(0);

            // accumulate clamped window from compacted LDS tile
#pragma unroll
            for (int k = 0; k < 32; ++k) {
                int v = k * 32 + lane;
                if (v < 1000) {
                    int c0 = v / 100, c1 = (v / 10) % 10, c2 = v % 10;
                    int u0 = c0 - off0, u1 = c1 - off1, u2 = c2 - off2;
                    if ((unsigned)u0 < (unsigned)a0 && (unsigned)u1 < (unsigned)a1 &&
                        (unsigned)u2 < (unsigned)a2)
                        acc[k] += tile[(u0 * a1 + u1) * a2 + u2];
                }
            }
        }
    }
#pragma unroll
    for (int k = 0; k < 32; ++k) {
        int v = k * 32 + lane;
        if (v < 1000) out[(size_t)r * 1000 + v] = acc[k];
    }
    if (lane == 0) {
        uloc[3 * r + 0] = cg0 * 10;
        uloc[3 * r + 1] = cg1 * 10;
        uloc[3 * r + 2] = cg2 * 10;
    }
}

// --------------------------------------------------------------------------
extern "C" void kernel_launch(void* const* d_in, const int* in_sizes, int n_in,
                              void* d_out, int out_size, void* d_ws, size_t ws_size,
                              hipStream_t stream) {
    const float* data = (const float*)d_in[0];
    const int*   loc  = (const int*)d_in[1];
    float* out  = (float*)d_out;
    int*   uloc = (int*)d_out + (size_t)M_CH * 1000;

    char* ws = (char*)d_ws;
    unsigned* keys  = (unsigned*)(ws);            // 65536 u32 (sorted in place)
    unsigned* ukeys = (unsigned*)(ws + 262144);   // 33600 u32
    unsigned* Uptr  = (unsigned*)(ws + 396544);   // 1 u32
    unsigned* mask2 = (unsigned*)(ws + 396560);   // 67200 u32 (8B aligned)

    kInit<<<(NSORT + 2 * M_CH + 255) / 256, 256, 0, stream>>>(loc, keys, mask2);

    kSortLocal<<<NSORT / 1024, 512, 0, stream>>>(keys);
    for (int k = 2048; k <= NSORT; k <<= 1) {
        for (int j = k >> 1; j >= 1024; j >>= 1)
            kMergeGlobal<<<NSORT / 256, 256, 0, stream>>>(keys, k, j);
        kMergeShared<<<NSORT / 1024, 512, 0, stream>>>(keys, k);
    }

    kDedup<<<1, 1024, 0, stream>>>(keys, ukeys, Uptr);
    kScatterMask<<<(M_CH + 255) / 256, 256, 0, stream>>>(loc, ukeys, Uptr, mask2);
    kGather<<<M_CH, 32, 0, stream>>>(data, loc, ukeys, Uptr,
                                     (const unsigned long long*)mask2, out, uloc);
}